// MultiPersonHeatmapOffsetAggregationLoss_26293789786918
// MI455X (gfx1250) — compile-verified
//
#include <hip/hip_runtime.h>
#include <hip/hip_bf16.h>
#include <math.h>

// ---------------------------------------------------------------------------
// MultiPersonHeatmapOffsetAggregationLoss for MI455X (gfx1250, wave32).
// Problem is ~1.5 MB of traffic -> pure latency bound; one fused kernel per
// (pose,keypoint) slice + a single-wave finisher that reduces the 255 partials
// with V_WMMA_F32_16X16X4_F32 (f32 matrix unit, same accumulation precision
// as the f32 reference).
// ---------------------------------------------------------------------------

typedef __attribute__((ext_vector_type(2))) float v2f;
typedef __attribute__((ext_vector_type(8))) float v8f;

#define NUM_KP   17
#define HH       33
#define WW       33
#define HW       (HH * WW)            // 1089
#define MAX_POSES 15
#define NBLK     (MAX_POSES * NUM_KP) // 255
#define RADIUS   3
#define THREADS  256

// One block per (pose, keypoint) slice. Produces two partial sums:
//   ws[blk]       = sum over pixels of BCE-with-logits vs binary disc target
//   ws[256 + blk] = sum over pixels*2 of masked SmoothL1 offset terms
__global__ void mphoal_partial_kernel(const float* __restrict__ pred_hm,   // (K,H,W)
                                      const float* __restrict__ tgt_hm,    // (P,K,H,W)
                                      const float* __restrict__ tkp,       // (P,K,2)
                                      const float* __restrict__ pred_off,  // (1,2K,H,W)
                                      float* __restrict__ ws) {
  const int blk = blockIdx.x;          // p*NUM_KP + k
  const int p   = blk / NUM_KP;
  const int k   = blk - p * NUM_KP;
  const int tid = threadIdx.x;
  (void)p;

  __shared__ float s_thr[HW];          // thresholded target heatmap (0/1)
  __shared__ float s_row[HW];          // horizontal 7-wide max
  __shared__ float s_red[2][THREADS];

  // Stage 1: threshold target heatmap into LDS.
  const float* hm = tgt_hm + (size_t)blk * HW;
  for (int i = tid; i < HW; i += THREADS)
    s_thr[i] = (hm[i] > 0.1f) ? 1.0f : 0.0f;
  __syncthreads();

  // Stage 2: separable max-pool, horizontal pass (pad region contributes 0).
  for (int i = tid; i < HW; i += THREADS) {
    const int y = i / WW;
    const int x = i - y * WW;
    const int x0 = (x - RADIUS < 0) ? 0 : x - RADIUS;
    const int x1 = (x + RADIUS > WW - 1) ? WW - 1 : x + RADIUS;
    float m = 0.0f;
    for (int xx2 = x0; xx2 <= x1; ++xx2) m = fmaxf(m, s_thr[y * WW + xx2]);
    s_row[i] = m;
  }
  __syncthreads();

  // Keypoint for this (p,k): x = [...,0], y = [...,1]; int() truncation.
  const float kx = tkp[(size_t)blk * 2 + 0];
  const float ky = tkp[(size_t)blk * 2 + 1];
  const int   xi = (int)truncf(kx);
  const int   yi = (int)truncf(ky);
  const bool  valid = ((kx != 0.0f && kx != -1.0f) || (ky != 0.0f && ky != -1.0f));

  const float* lg  = pred_hm  + (size_t)k * HW;
  const float* po0 = pred_off + (size_t)(2 * k)     * HW;
  const float* po1 = pred_off + (size_t)(2 * k + 1) * HW;

  float bce_sum = 0.0f;
  float off_sum = 0.0f;
  for (int i = tid; i < HW; i += THREADS) {
    const int y = i / WW;
    const int x = i - y * WW;

    // --- BCE with logits vs binary disc target ---
    const float l  = lg[i];
    const int dy = y - yi, dx = x - xi;
    const float b = (valid && (dy * dy + dx * dx <= RADIUS * RADIUS)) ? 1.0f : 0.0f;
    bce_sum += fmaxf(l, 0.0f) - l * b + log1pf(expf(-fabsf(l)));

    // --- mask: vertical pass of separable 7x7 max-pool ---
    const int y0 = (y - RADIUS < 0) ? 0 : y - RADIUS;
    const int y1 = (y + RADIUS > HH - 1) ? HH - 1 : y + RADIUS;
    float m = 0.0f;
    for (int yy2 = y0; yy2 <= y1; ++yy2) m = fmaxf(m, s_row[yy2 * WW + x]);

    // --- masked SmoothL1 (beta = 1). Reference gt pairing:
    //     comp0 = kp_x - 8*y ; comp1 = kp_y - 8*x  (replicated exactly). ---
    if (m > 0.0f) {
      const float d0 = po0[i] - (kx - 8.0f * (float)y);
      const float d1 = po1[i] - (ky - 8.0f * (float)x);
      const float a0 = fabsf(d0), a1 = fabsf(d1);
      off_sum += (a0 < 1.0f) ? 0.5f * d0 * d0 : a0 - 0.5f;
      off_sum += (a1 < 1.0f) ? 0.5f * d1 * d1 : a1 - 0.5f;
    }
  }

  // Block tree reduction (two streams).
  s_red[0][tid] = bce_sum;
  s_red[1][tid] = off_sum;
  __syncthreads();
  for (int s = THREADS / 2; s > 0; s >>= 1) {
    if (tid < s) {
      s_red[0][tid] += s_red[0][tid + s];
      s_red[1][tid] += s_red[1][tid + s];
    }
    __syncthreads();
  }
  if (tid == 0) {
    ws[blk]       = s_red[0][0];
    ws[256 + blk] = s_red[1][0];
  }
}

// Single-wave finisher: weights + scales the 255 partials, then reduces them
// on the f32 matrix unit with V_WMMA_F32_16X16X4_F32 (ones as B-matrix:
// D[m,n] = sum_k A[m,k] + C[m,n]; grand total = column 0 of C, i.e. lane 0's
// eight accumulators (M=0..7) plus lane 16's (M=8..15)).
__global__ void mphoal_finish_kernel(const float* __restrict__ tkp,
                                     const float* __restrict__ ws,
                                     float* __restrict__ out) {
  const int lane = threadIdx.x;  // 0..31 (wave32)

  // num_people: pose counts if ANY of its 34 keypoint coords != -1.
  int np = 0;
  for (int p = 0; p < MAX_POSES; ++p) {
    bool v = false;
    for (int i = 0; i < NUM_KP * 2; ++i)
      v = v || (tkp[p * NUM_KP * 2 + i] != -1.0f);
    np += v ? 1 : 0;
  }

  const float w_bce = 4.0f / (float)(NUM_KP * HW);      // heatmap_weight / (K*h*w)
  const float w_off = 1.0f / (float)(NUM_KP * HW * 2);  // offset_weight  / (K*h*w*2)

  // Pack 256 (padded) combined values into 4 WMMA iterations, 2 per lane each.
  float vals[8];
  for (int t = 0; t < 4; ++t) {
    for (int j = 0; j < 2; ++j) {
      const int idx = t * 64 + lane * 2 + j;
      float v = 0.0f;
      if (idx < NBLK) {
        const int p = idx / NUM_KP;
        const float inloop = (p < np) ? 1.0f : 0.0f;  // arange(P) < num_people
        v = inloop * (ws[idx] * w_bce + ws[256 + idx] * w_off);
      }
      vals[t * 2 + j] = v;
    }
  }

  v8f c = {};               // f32 accumulator tile
  v2f bones;
  bones.x = 1.0f;
  bones.y = 1.0f;
  for (int t = 0; t < 4; ++t) {
    v2f a;
    a.x = vals[2 * t];
    a.y = vals[2 * t + 1];
    // 8 args: (neg_a, A, neg_b, B, c_mod, C, reuse_a, reuse_b)
    c = __builtin_amdgcn_wmma_f32_16x16x4_f32(false, a, false, bones,
                                              (short)0, c, false, false);
  }

  const float s = c[0] + c[1] + c[2] + c[3] + c[4] + c[5] + c[6] + c[7];
  const float total = __shfl(s, 0, 32) + __shfl(s, 16, 32);

  if (lane == 0) out[0] = total / (float)np;
}

extern "C" void kernel_launch(void* const* d_in, const int* in_sizes, int n_in,
                              void* d_out, int out_size, void* d_ws, size_t ws_size,
                              hipStream_t stream) {
  (void)in_sizes; (void)n_in; (void)out_size; (void)ws_size;
  const float* pred_hm  = (const float*)d_in[0];  // (17,33,33)
  const float* tgt_hm   = (const float*)d_in[1];  // (15,17,33,33)
  const float* tkp      = (const float*)d_in[2];  // (15,17,2)
  const float* pred_off = (const float*)d_in[3];  // (1,34,33,33)
  // d_in[4] target_offsets: unused by the forward pass.
  // d_in[5] max_num_poses: compile-time constant (15).

  float* ws = (float*)d_ws;   // 512 floats of partials

  mphoal_partial_kernel<<<NBLK, THREADS, 0, stream>>>(pred_hm, tgt_hm, tkp, pred_off, ws);
  mphoal_finish_kernel<<<1, 32, 0, stream>>>(tkp, ws, (float*)d_out);
}